// NeuralRegisterIndexer_18975165514077
// MI455X (gfx1250) — compile-verified
//
#include <hip/hip_runtime.h>
#include <math.h>

typedef __attribute__((ext_vector_type(2))) float v2f;
typedef __attribute__((ext_vector_type(4))) float v4f;
typedef __attribute__((ext_vector_type(8))) float v8f;

#define KD   128   // KEY_DIM
#define NREG 32    // N_REGS
#define BW   64    // BIT_WIDTH

// ---------------------------------------------------------------------------
// Step 1: h = gelu(bits @ W1 + b1), 32 x 128.  bits are 0/1 -> row selection.
// 4096 threads total.
// ---------------------------------------------------------------------------
__global__ void encode_gelu_kernel(const float* __restrict__ W1,
                                   const float* __restrict__ b1,
                                   float* __restrict__ h) {
    int t = blockIdx.x * blockDim.x + threadIdx.x;   // 0..4095
    int r = t >> 7;       // register id 0..31
    int c = t & 127;      // key-dim column
    float s = b1[c];
#pragma unroll
    for (int j = 0; j < 5; ++j)
        if ((r >> j) & 1) s += W1[j * KD + c];
    // exact GELU: 0.5*x*(1+erf(x/sqrt(2)))
    float g = 0.5f * s * (1.0f + erff(s * 0.70710678118654752440f));
    h[r * KD + c] = g;
}

// ---------------------------------------------------------------------------
// Generic tiny f32 GEMM with V_WMMA_F32_16X16X4_F32.
// One wave (32 threads, EXEC all ones) per 16x16 C tile; K stepped by 4.
// A: MxK row-major.  TRANS_B=0: B is KxN row-major.  TRANS_B=1: B is NxK
// row-major (i.e. we multiply by B^T).  Optional bias added per column.
//
// Fragment layouts (CDNA5 ISA 7.12.2, wave32):
//   A 16x4 f32 : lane<16 -> {K=k,k+1} of row M=lane ; lane>=16 -> {K=k+2,k+3}
//   B  4x16 f32: mirror of A with M->N (B column-major per lane)
//   C/D 16x16  : VGPR r -> row r (lanes 0-15) / row r+8 (lanes 16-31), N=lane&15
// ---------------------------------------------------------------------------
template <bool TRANS_B>
__global__ void wmma_gemm_f32(const float* __restrict__ A,
                              const float* __restrict__ Bm,
                              const float* __restrict__ bias,   // may be null
                              float* __restrict__ C,
                              int M, int N, int K) {
    const int lane = threadIdx.x;      // 0..31
    const int half = lane >> 4;        // 0 or 1
    const int lid  = lane & 15;
    const int mtiles = M >> 4;
    const int mt = blockIdx.x % mtiles;
    const int nt = blockIdx.x / mtiles;
    const int m0 = mt << 4;
    const int n0 = nt << 4;

    v8f acc = {};
    for (int k = 0; k < K; k += 4) {
        const int ka = k + 2 * half;
        v2f a, b;
        a.x = A[(m0 + lid) * K + ka];
        a.y = A[(m0 + lid) * K + ka + 1];
        if (TRANS_B) {
            b.x = Bm[(n0 + lid) * K + ka];
            b.y = Bm[(n0 + lid) * K + ka + 1];
        } else {
            b.x = Bm[(ka    ) * N + n0 + lid];
            b.y = Bm[(ka + 1) * N + n0 + lid];
        }
        // 8 args: (neg_a, A, neg_b, B, c_mod, C, reuse_a, reuse_b)
        acc = __builtin_amdgcn_wmma_f32_16x16x4_f32(
            false, a, false, b, (short)0, acc, false, false);
    }
    const float bn = bias ? bias[n0 + lid] : 0.0f;
#pragma unroll
    for (int r = 0; r < 8; ++r)
        C[(m0 + r + 8 * half) * N + n0 + lid] = acc[r] + bn;
}

// ---------------------------------------------------------------------------
// Step 4: row-wise softmax over 32 registers (one thread per row, exact expf).
// ---------------------------------------------------------------------------
__global__ void softmax_kernel(const float* __restrict__ sim,
                               const float* __restrict__ temp_p,
                               float* __restrict__ attn) {
    const int r = threadIdx.x;                 // 0..31
    const float inv = 1.0f / fmaxf(fabsf(temp_p[0]), 0.1f);
    float mx = -INFINITY;
#pragma unroll
    for (int c = 0; c < NREG; ++c) mx = fmaxf(mx, sim[r * NREG + c] * inv);
    float e[NREG];
    float sum = 0.0f;
#pragma unroll
    for (int c = 0; c < NREG; ++c) {
        e[c] = expf(sim[r * NREG + c] * inv - mx);
        sum += e[c];
    }
    const float rs = 1.0f / sum;
#pragma unroll
    for (int c = 0; c < NREG; ++c) attn[r * NREG + c] = e[c] * rs;
}

// ---------------------------------------------------------------------------
// Step 6: gather. out[b, :] = (idx[b]==31) ? 0 : table[idx[b], :].
// 16 bytes per thread, fully coalesced, nontemporal streaming stores
// (256 MB output won't fit/benefit from the 192 MB L2).
// ---------------------------------------------------------------------------
__global__ void gather_kernel(const int* __restrict__ idx,
                              const v4f* __restrict__ table4,   // 32*16 v4f
                              v4f* __restrict__ out4) {
    const long long t = (long long)blockIdx.x * blockDim.x + threadIdx.x;
    const int row = (int)(t >> 4);
    const int q   = (int)(t & 15);
    const int id  = idx[row];
    v4f v = table4[id * 16 + q];
    const float m = (id == 31) ? 0.0f : 1.0f;
    v *= m;
    __builtin_nontemporal_store(v, &out4[t]);
}

// ---------------------------------------------------------------------------
extern "C" void kernel_launch(void* const* d_in, const int* in_sizes, int n_in,
                              void* d_out, int out_size, void* d_ws, size_t ws_size,
                              hipStream_t stream) {
    const int*   idx  = (const int*)  d_in[0];
    const float* W1   = (const float*)d_in[1];
    const float* b1   = (const float*)d_in[2];
    const float* W2   = (const float*)d_in[3];
    const float* b2   = (const float*)d_in[4];
    const float* keys = (const float*)d_in[5];
    const float* temp = (const float*)d_in[6];
    const float* rv   = (const float*)d_in[7];
    float* out = (float*)d_out;

    // Workspace layout (floats): h[4096] query[4096] sim[1024] attn[1024] table[2048]
    float* ws    = (float*)d_ws;
    float* h     = ws;
    float* query = ws + 4096;
    float* sim   = ws + 8192;
    float* attn  = ws + 9216;
    float* table = ws + 10240;

    const int nB = in_sizes[0];                 // 1048576 rows

    // 1) h = gelu(bits @ W1 + b1)                      [32 x 128]
    encode_gelu_kernel<<<16, 256, 0, stream>>>(W1, b1, h);
    // 2) query = h @ W2 + b2                           [32 x 128], K=128 -> 2x8 tiles
    wmma_gemm_f32<false><<<16, 32, 0, stream>>>(h, W2, b2, query, 32, KD, KD);
    // 3) sim = query @ keys^T                          [32 x 32], K=128 -> 2x2 tiles
    wmma_gemm_f32<true ><<< 4, 32, 0, stream>>>(query, keys, nullptr, sim, 32, NREG, KD);
    // 4) attn = softmax(sim / temp)
    softmax_kernel<<<1, 32, 0, stream>>>(sim, temp, attn);
    // 5) table = attn @ register_values                [32 x 64], K=32 -> 2x4 tiles
    wmma_gemm_f32<false><<< 8, 32, 0, stream>>>(attn, rv, nullptr, table, 32, BW, NREG);
    // 6) out[b,:] = table[idx[b],:] (masked for XZR)   256 MB streaming write
    const long long nthreads = (long long)nB * 16;    // one v4f per thread
    gather_kernel<<<(unsigned)(nthreads / 256), 256, 0, stream>>>(
        idx, (const v4f*)table, (v4f*)out);
}